// AttentionBasedTimestamps_58050777972825
// MI455X (gfx1250) — compile-verified
//
#include <hip/hip_runtime.h>
#include <hip/hip_bf16.h>

// ---------------------------------------------------------------------------
// AttentionBasedTimestamps for MI455X (gfx1250, wave32)
//
// Pass 1: mean over 64 (L,H) planes of the sliced region, expressed as
//         V_WMMA_F32_16X16X4_F32 with an all-ones B matrix (K = 4 planes per
//         WMMA, 16 accumulating WMMAs = 64 planes).  HBM-bandwidth bound:
//         reads exactly the 235 MB slice, writes 3.7 MB to workspace.
// Pass 2: per-row stats (max/argmax, half-max span, softmax entropy) with
//         one wave per 512-element row, shfl_xor wave reductions.  L2-resident.
// ---------------------------------------------------------------------------

typedef __attribute__((ext_vector_type(2))) float v2f;
typedef __attribute__((ext_vector_type(8))) float v8f;

#define LL 4
#define BB 4
#define HH 16
#define SS 1024
#define TT 448               // SS - TEXT_START(576)
#define FF 512               // AUDIO_END(576) - AUDIO_START(64)
#define ROW0 576
#define COL0 64
#define FRAME_MS_C 40.0f

__global__ __launch_bounds__(256)
void abt_reduce_mean_wmma(const float* __restrict__ attn, float* __restrict__ w)
{
    const int lane = threadIdx.x & 31;
    const int wave = threadIdx.x >> 5;
    const int tile = blockIdx.x * 8 + wave;     // BB*TT*(FF/16) = 57344 tiles

    const int ftile = tile & 31;                // 32 f-tiles per row
    const int rowid = tile >> 5;                // b*TT + t
    const int t     = rowid % TT;
    const int b     = rowid / TT;
    const int f0    = ftile << 4;

    // A-matrix (16x4 fp32) layout: lanes 0-15 hold M=lane with K0 (vgpr0) and
    // K1 (vgpr1); lanes 16-31 hold M=lane-16 with K2/K3.  M = f within tile,
    // K = attention plane p = l*16 + h.
    const int rowA = lane & 15;
    const int koff = (lane < 16) ? 0 : 2;

    const size_t planeH = (size_t)SS * SS;            // h stride (1 Mi elems)
    const size_t planeL = (size_t)BB * HH * planeH;   // l stride (64 Mi elems)
    const size_t base   = (size_t)b * HH * planeH
                        + (size_t)(ROW0 + t) * SS
                        + (size_t)(COL0 + f0 + rowA);

    v8f acc = {0.f, 0.f, 0.f, 0.f, 0.f, 0.f, 0.f, 0.f};
    v2f bones; bones.x = 1.0f; bones.y = 1.0f;        // B = all ones (4x16)

#pragma unroll
    for (int g = 0; g < 16; ++g) {
        const int p0 = 4 * g + koff;                  // plane for A vgpr0
        const int p1 = p0 + 1;                        // plane for A vgpr1
        const size_t off0 = (size_t)(p0 >> 4) * planeL + (size_t)(p0 & 15) * planeH;
        const size_t off1 = (size_t)(p1 >> 4) * planeL + (size_t)(p1 & 15) * planeH;
        v2f a;
        a.x = attn[off0 + base];                      // coalesced 64B per half-wave
        a.y = attn[off1 + base];
        // D[m,n] = sum_k A[m,k] * 1 + C[m,n]  -> accumulates 4 planes per call
        acc = __builtin_amdgcn_wmma_f32_16x16x4_f32(
                  false, a, false, bones, (short)0, acc, false, false);
    }

    // Every D column is identical, so lane 0 holds row-sums for M=0..7 and
    // lane 16 holds M=8..15.  Two b128 stores per active lane.
    if ((lane & 15) == 0) {
        const float s = 1.0f / 64.0f;
        float* dst = w + (size_t)rowid * FF + f0 + ((lane >> 4) << 3);
        float4 lo = make_float4(acc[0] * s, acc[1] * s, acc[2] * s, acc[3] * s);
        float4 hi = make_float4(acc[4] * s, acc[5] * s, acc[6] * s, acc[7] * s);
        *reinterpret_cast<float4*>(dst)     = lo;
        *reinterpret_cast<float4*>(dst + 4) = hi;
    }
}

__global__ __launch_bounds__(256)
void abt_row_stats(const float* __restrict__ w, float* __restrict__ out)
{
    const int lane = threadIdx.x & 31;
    const int wave = threadIdx.x >> 5;
    const int row  = blockIdx.x * 8 + wave;           // 0 .. BB*TT-1
    const float* wr = w + (size_t)row * FF;

    float vals[16];
#pragma unroll
    for (int i = 0; i < 16; ++i) vals[i] = wr[lane + 32 * i];

    // --- max + first-occurrence argmax (jnp.argmax tie-break: lowest index) ---
    float mval = vals[0];
    int   midx = lane;
#pragma unroll
    for (int i = 1; i < 16; ++i) {
        const int f = lane + 32 * i;
        if (vals[i] > mval) { mval = vals[i]; midx = f; }
    }
#pragma unroll
    for (int s = 16; s >= 1; s >>= 1) {
        const float ov = __shfl_xor(mval, s, 32);
        const int   oi = __shfl_xor(midx, s, 32);
        if (ov > mval || (ov == mval && oi < midx)) { mval = ov; midx = oi; }
    }

    // --- half-max span + fused softmax-entropy sums ---
    const float thresh = 0.5f * mval;
    int   firstI = 0x7FFFFFFF;
    int   lastI  = -1;
    float s1 = 0.0f;   // sum exp(x - m)
    float s2 = 0.0f;   // sum (x - m) * exp(x - m)
#pragma unroll
    for (int i = 0; i < 16; ++i) {
        const int   f = lane + 32 * i;
        const float x = vals[i] - mval;
        const float e = __expf(x);
        s1 += e;
        s2 += x * e;
        if (vals[i] > thresh) {
            firstI = (f < firstI) ? f : firstI;
            lastI  = (f > lastI)  ? f : lastI;
        }
    }
#pragma unroll
    for (int s = 16; s >= 1; s >>= 1) {
        s1 += __shfl_xor(s1, s, 32);
        s2 += __shfl_xor(s2, s, 32);
        const int of = __shfl_xor(firstI, s, 32);
        const int ol = __shfl_xor(lastI,  s, 32);
        firstI = (of < firstI) ? of : firstI;
        lastI  = (ol > lastI)  ? ol : lastI;
    }

    if (lane == 0) {
        const bool has   = (lastI >= 0);
        const int startF = has ? firstI : midx;
        const int endF   = has ? lastI  : midx;
        // entropy = log(Z) - E[(x-m)] under softmax; confidence = 1 - H/log(F)
        const float entropy = __logf(s1) - s2 / s1;
        const float conf    = 1.0f - entropy * (1.0f / 6.2383246250f); // /log(512)
        const int N = BB * TT;
        out[0 * N + row] = (float)startF;
        out[1 * N + row] = (float)endF;
        out[2 * N + row] = (float)startF * FRAME_MS_C;
        out[3 * N + row] = (float)endF   * FRAME_MS_C;
        out[4 * N + row] = conf;
    }
}

extern "C" void kernel_launch(void* const* d_in, const int* in_sizes, int n_in,
                              void* d_out, int out_size, void* d_ws, size_t ws_size,
                              hipStream_t stream)
{
    (void)in_sizes; (void)n_in; (void)out_size; (void)ws_size;
    const float* attn = (const float*)d_in[0];
    float*       w    = (float*)d_ws;     // BB*TT*FF fp32 = 3.67 MB scratch
    float*       out  = (float*)d_out;    // 5 * BB*TT floats

    const int tiles = BB * TT * (FF / 16);            // 57344, 8 waves/block
    abt_reduce_mean_wmma<<<tiles / 8, 256, 0, stream>>>(attn, w);

    const int rows = BB * TT;                          // 1792, 8 waves/block
    abt_row_stats<<<rows / 8, 256, 0, stream>>>(w, out);
}